// FOAMEmbedding_21414706938520
// MI455X (gfx1250) — compile-verified
//
#include <hip/hip_runtime.h>
#include <hip/hip_bf16.h>

// FOAM-style equivariant embedding for MI455X (gfx1250, wave32).
//
// Edge phase: ~180MB traffic (~8us @ 23.3TB/s), segment-sum with no atomics
// (edge_src sorted). Contraction: 8.85 GFLOP fp32 via V_WMMA_F32_16X16X4_F32.
// W is pre-swizzled in global memory into a k-pair-interleaved layout so each
// B fragment is a single contiguous ds_load_b64; A-tile (32 nodes, 144KB) and
// swizzled W_l (128KB) are staged in LDS via GLOBAL_LOAD_ASYNC_TO_LDS_B128
// (ASYNCcnt). Two node tiles per wave -> each B fragment feeds 2 WMMAs:
// per k-step = 6 ds loads + 8 wmma.

typedef __attribute__((ext_vector_type(2))) float v2f;
typedef __attribute__((ext_vector_type(8))) float v8f;
typedef __attribute__((address_space(3))) const float lds_cfloat;
typedef __attribute__((address_space(3))) const v2f   lds_cv2f;

#define NCH        128            // N_RADIAL * N_SPEC = 8*16
#define OUTSTRIDE  528            // 16 + 128 + 3*128
#define CUTOFF_F   5.0f
#define NODES_PER_BLOCK 32
#define WFLOATS    32768          // 128*256 per l
#define LDS_A_BYTES 147456        // 32 nodes * 9 m * 128 ch * 4B
#define LDS_W_BYTES 131072        // 128 * 256 * 4B
#define LDS_TOTAL  (LDS_A_BYTES + LDS_W_BYTES)   // 272 KB < 320 KB/WGP

// ---------------------------------------------------------------------------
// Kernel A: per-node edge range via lower_bound over sorted edge_src.
// ---------------------------------------------------------------------------
__global__ void k_node_offsets(const int* __restrict__ src, int* __restrict__ off,
                               int nNodes, int nEdges) {
  int n = blockIdx.x * blockDim.x + threadIdx.x;
  if (n > nNodes) return;
  if (n == nNodes) { off[n] = nEdges; return; }
  int lo = 0, hi = nEdges;
  while (lo < hi) {
    int mid = (lo + hi) >> 1;
    if (src[mid] < n) lo = mid + 1; else hi = mid;
  }
  off[n] = lo;
}

// ---------------------------------------------------------------------------
// Kernel W: one-time swizzle of W0/W1/W2 into k-pair-interleaved layout:
//   wp[l][(k>>1)*512 + c*2 + (k&1)] = W_l[k][c]
// so B fragments (rows k', k'+1 at column c) are contiguous 8 bytes.
// ---------------------------------------------------------------------------
__global__ void k_w_swizzle(const float* __restrict__ W0, const float* __restrict__ W1,
                            const float* __restrict__ W2, float* __restrict__ wp) {
  const int o = blockIdx.x * blockDim.x + threadIdx.x;   // 0 .. 3*32768-1
  const int l = o >> 15;
  const int i = o & (WFLOATS - 1);
  const float* W = (l == 0) ? W0 : (l == 1) ? W1 : W2;
  const int parity = i & 1;
  const int c      = (i >> 1) & 255;
  const int kpair  = i >> 9;
  wp[o] = W[(kpair * 2 + parity) * 256 + c];
}

// ---------------------------------------------------------------------------
// Kernel B: one 128-thread block per (padded) node; thread c owns channel
// c = radial*16 + spec, accumulates rhoi[n][0..8][c] over the node's edge
// segment. Pad nodes write zero rhoi rows so WMMA tiles are safe.
// ---------------------------------------------------------------------------
__global__ void __launch_bounds__(128)
k_edge_accum(const float* __restrict__ dist, const float* __restrict__ vec,
             const float* __restrict__ sw, const float* __restrict__ stab,
             const int* __restrict__ species, const int* __restrict__ edst,
             const int* __restrict__ off, float* __restrict__ rhoi,
             float* __restrict__ out, int nNodes) {
  const int n = blockIdx.x;
  const int c = threadIdx.x;      // 0..127
  const int rIdx = c >> 4;        // radial 0..7
  const int sIdx = c & 15;        // species feature 0..15

  float acc[9];
#pragma unroll
  for (int m = 0; m < 9; ++m) acc[m] = 0.0f;

  if (n < nNodes) {
    const int e0 = off[n], e1 = off[n + 1];
    const float kco  = 3.14159265358979323846f / CUTOFF_F * (float)(rIdx + 1);
    const float norm = 0.6324555320336759f;   // sqrt(2/5)
    const float s3  = 1.7320508075688772f;
    const float s5  = 2.23606797749979f;
    const float s15 = 3.872983346207417f;
    for (int e = e0; e < e1; ++e) {
      const float r   = dist[e];
      const float inv = 1.0f / r;
      const float rbv = norm * __sinf(kco * r) * inv * sw[e];
      const float x = vec[3 * e + 0] * inv;
      const float y = vec[3 * e + 1] * inv;
      const float z = vec[3 * e + 2] * inv;
      const float d = rbv * stab[species[edst[e]] * 16 + sIdx];
      acc[0] += d;
      acc[1] += s3 * x * d;
      acc[2] += s3 * y * d;
      acc[3] += s3 * z * d;
      acc[4] += s15 * x * y * d;
      acc[5] += s15 * y * z * d;
      acc[6] += 0.5f * s5 * (3.0f * z * z - 1.0f) * d;
      acc[7] += s15 * x * z * d;
      acc[8] += 0.5f * s15 * (x * x - y * y) * d;
    }
  }

  float* rp = rhoi + (size_t)n * 9 * NCH + c;
#pragma unroll
  for (int m = 0; m < 9; ++m) rp[m * NCH] = acc[m];

  if (n < nNodes) {
    out[(size_t)n * OUTSTRIDE + 16 + c] = acc[0];             // rhoi[:,0,:]
    if (c < 16) out[(size_t)n * OUTSTRIDE + c] = stab[species[n] * 16 + c]; // senc
  }
}

// ---------------------------------------------------------------------------
// Async copy helpers: GLOBAL_LOAD_ASYNC_TO_LDS_B128 (per-lane, ASYNCcnt).
// LDS destination address = low 32 bits of the generic __shared__ pointer.
// ---------------------------------------------------------------------------
__device__ __forceinline__ void async_copy_b128(unsigned lds_off, const void* gaddr) {
  asm volatile("global_load_async_to_lds_b128 %0, %1, off"
               :: "v"(lds_off), "v"(gaddr) : "memory");
}
__device__ __forceinline__ void wait_async0() {
  asm volatile("s_wait_asynccnt 0x0" ::: "memory");
}

// ---------------------------------------------------------------------------
// Kernel C: xi_l[n,c] = (1/sqrt(2l+1)) * sum_m (rhoi_l[n,m,:]@W[:,c]) *
//                                              (rhoi_l[n,m,:]@W[:,c+128])
// Block = 32 nodes (2 M-tiles), 4 waves. Wave w owns column tiles
// {2w,2w+1,2w+8,2w+9}; pairing (j, j+8) keeps xl*yl lane-local. Each B
// fragment is shared by both M-tiles -> 8 wmma per k-step. B fragments are
// single ds_load_b64s thanks to the k-pair-interleaved W layout.
//
// V_WMMA_F32_16X16X4_F32 fragment layouts (ISA 7.12.2):
//   A 16x4 : lane L -> M=L%16, VGPR{0,1} = K = 2*(L/16) + {0,1}
//   B 4x16 : lane L -> N=L%16, VGPR{0,1} = K = 2*(L/16) + {0,1}
//   C 16x16: lane L, VGPR r -> M = r + 8*(L/16), N = L%16
// ---------------------------------------------------------------------------
__global__ void __launch_bounds__(128)
k_contract_wmma(const float* __restrict__ rhoi, const float* __restrict__ wp,
                float* __restrict__ out, int nNodes) {
  extern __shared__ char lds_raw[];
  float* aT = (float*)lds_raw;                       // [32 nodes][9 m][128 ch]
  float* wT = (float*)(lds_raw + LDS_A_BYTES);       // [64 kpair][256 col] x v2f
  const unsigned ldsA = (unsigned)(uintptr_t)aT;
  const unsigned ldsW = (unsigned)(uintptr_t)wT;

  const int tid   = threadIdx.x;
  const int lane  = tid & 31;
  const int wave  = tid >> 5;              // 0..3
  const int node0 = blockIdx.x * NODES_PER_BLOCK;
  const int lm    = lane & 15;             // A: M row; B/C: N col
  const int half  = lane >> 4;             // lane half
  const int khalf = half * 2;              // K base for this lane
  const int ct0   = 2 * wave;              // first owned column tile

  // --- stage A tile (contiguous 147456 bytes at rhoi[node0][0][0]) ---
  {
    const char* gA = (const char*)(rhoi + (size_t)node0 * 9 * NCH);
    for (int i = tid * 16; i < LDS_A_BYTES; i += 128 * 16)
      async_copy_b128(ldsA + i, gA + i);
  }

  const int    mlo[3] = {0, 1, 4};
  const int    mhi[3] = {1, 4, 9};
  const float  scl[3] = {1.0f, 0.5773502691896258f, 0.4472135954999579f};

  // per-lane B base (bytes): float2 index = (kpair=half)*256 + ct0*16 + lm
  const unsigned wbase = ldsW + (unsigned)(half * 256 + ct0 * 16 + lm) * 8u;

  for (int l = 0; l < 3; ++l) {
    __syncthreads();                        // previous-l consumers done
    {                                       // stage swizzled W_l (131072 bytes)
      const char* gW = (const char*)(wp + (size_t)l * WFLOATS);
      for (int i = tid * 16; i < LDS_W_BYTES; i += 128 * 16)
        async_copy_b128(ldsW + i, gW + i);
    }
    wait_async0();                          // own async copies complete
    __syncthreads();                        // everyone's copies visible

    v8f p00 = {}, p01 = {}, p10 = {}, p11 = {};
    for (int m = mlo[l]; m < mhi[l]; ++m) {
      v8f c00 = {}, c01 = {}, c02 = {}, c03 = {};   // M-tile 0, col tiles
      v8f c10 = {}, c11 = {}, c12 = {}, c13 = {};   // M-tile 1
      const float* ar0 = aT + (lm * 9 + m) * NCH + khalf;
      const float* ar1 = ar0 + 16 * 9 * NCH;        // node lm+16
      unsigned wofs = wbase;                        // running B address (bytes)
#pragma unroll 4
      for (int k = 0; k < NCH; k += 4) {
        asm volatile("" : "+v"(wofs));              // keep offset opaque ->
        v2f aA; aA[0] = ar0[k];   aA[1] = ar0[k + 1];      // ds 2addr
        v2f aB; aB[0] = ar1[k];   aB[1] = ar1[k + 1];
        v2f b0 = *(lds_cv2f*)(uintptr_t)(wofs);            // col ct0+0 (b64)
        v2f b1 = *(lds_cv2f*)(uintptr_t)(wofs + 128u);     // col ct0+1
        v2f b2 = *(lds_cv2f*)(uintptr_t)(wofs + 1024u);    // col ct0+8
        v2f b3 = *(lds_cv2f*)(uintptr_t)(wofs + 1152u);    // col ct0+9
        c00 = __builtin_amdgcn_wmma_f32_16x16x4_f32(false, aA, false, b0, (short)0, c00, false, false);
        c10 = __builtin_amdgcn_wmma_f32_16x16x4_f32(false, aB, false, b0, (short)0, c10, false, false);
        c01 = __builtin_amdgcn_wmma_f32_16x16x4_f32(false, aA, false, b1, (short)0, c01, false, false);
        c11 = __builtin_amdgcn_wmma_f32_16x16x4_f32(false, aB, false, b1, (short)0, c11, false, false);
        c02 = __builtin_amdgcn_wmma_f32_16x16x4_f32(false, aA, false, b2, (short)0, c02, false, false);
        c12 = __builtin_amdgcn_wmma_f32_16x16x4_f32(false, aB, false, b2, (short)0, c12, false, false);
        c03 = __builtin_amdgcn_wmma_f32_16x16x4_f32(false, aA, false, b3, (short)0, c03, false, false);
        c13 = __builtin_amdgcn_wmma_f32_16x16x4_f32(false, aB, false, b3, (short)0, c13, false, false);
        wofs += 4096u;                              // 2 kpairs * 256 cols * 8B
      }
#pragma unroll
      for (int r = 0; r < 8; ++r) {
        p00[r] += c00[r] * c02[r];            // xl*yl: cols j and j+8
        p01[r] += c01[r] * c03[r];
        p10[r] += c10[r] * c12[r];
        p11[r] += c11[r] * c13[r];
      }
    }
    const int chBase = 144 + l * NCH;         // 16 + 128 + l*128
#pragma unroll
    for (int r = 0; r < 8; ++r) {
      const int nA = node0 + r + half * 8;          // M-tile 0 row
      const int nB = nA + 16;                       // M-tile 1 row
      if (nA < nNodes) {
        float* o = out + (size_t)nA * OUTSTRIDE + chBase + ct0 * 16 + lm;
        o[0]  = p00[r] * scl[l];
        o[16] = p01[r] * scl[l];
      }
      if (nB < nNodes) {
        float* o = out + (size_t)nB * OUTSTRIDE + chBase + ct0 * 16 + lm;
        o[0]  = p10[r] * scl[l];
        o[16] = p11[r] * scl[l];
      }
    }
  }
}

// ---------------------------------------------------------------------------
extern "C" void kernel_launch(void* const* d_in, const int* in_sizes, int n_in,
                              void* d_out, int out_size, void* d_ws, size_t ws_size,
                              hipStream_t stream) {
  const float* dist    = (const float*)d_in[0];
  const float* vec     = (const float*)d_in[1];
  const float* sw      = (const float*)d_in[2];
  const float* stab    = (const float*)d_in[3];
  const float* W0      = (const float*)d_in[4];
  const float* W1      = (const float*)d_in[5];
  const float* W2      = (const float*)d_in[6];
  const int*   species = (const int*)d_in[7];
  const int*   esrc    = (const int*)d_in[8];
  const int*   edst    = (const int*)d_in[9];
  float* out = (float*)d_out;

  const int nNodes = in_sizes[7];
  const int nEdges = in_sizes[0];
  const int nPad   = ((nNodes + NODES_PER_BLOCK - 1) / NODES_PER_BLOCK) * NODES_PER_BLOCK;

  // workspace: [ offsets (nNodes+1 ints) | 256B pad | rhoi | wp (3*32768 f32) ]
  int*   off  = (int*)d_ws;
  size_t roff = (((size_t)(nNodes + 1) * sizeof(int)) + 255) & ~(size_t)255;
  float* rhoi = (float*)((char*)d_ws + roff);
  float* wpck = rhoi + (size_t)nPad * 9 * NCH;

  k_node_offsets<<<(nNodes + 256) / 256, 256, 0, stream>>>(esrc, off, nNodes, nEdges);
  k_w_swizzle<<<(3 * WFLOATS) / 256, 256, 0, stream>>>(W0, W1, W2, wpck);
  k_edge_accum<<<nPad, 128, 0, stream>>>(dist, vec, sw, stab, species, edst,
                                         off, rhoi, out, nNodes);
  k_contract_wmma<<<nPad / NODES_PER_BLOCK, 128, LDS_TOTAL, stream>>>(rhoi, wpck, out, nNodes);
}